// RotatedCIoULoss_63591285785212
// MI455X (gfx1250) — compile-verified
//
#include <hip/hip_runtime.h>
#include <math.h>

// CDNA5 (gfx1250) ext-vector types for the fp32 WMMA path.
typedef __attribute__((ext_vector_type(2))) float v2f_t;
typedef __attribute__((ext_vector_type(8))) float v8f_t;

// ---------------------------------------------------------------------------
// Wave32 sum reduction using V_WMMA_F32_16X16X4_F32.
// A = 16x4 ones, B = lane partials (VGPR0 = s, VGPR1 = 0).
// D[m,n] = sum_k B[k,n]  -> every row of D holds the 16 column sums of B.
// Sum of all 16 columns == sum of all 64 B elements == sum of the 32 lane
// values. c[0] holds row M=0 (lanes 0-15) / M=8 (lanes 16-31); both rows are
// identical, so 4 xor-shuffles within each 16-lane half yield the wave total
// in every lane. EXEC must be all ones here (guaranteed: called after all
// divergent loops reconverge, full 256-thread blocks).
// ---------------------------------------------------------------------------
__device__ __forceinline__ float wave_sum_wmma(float s) {
  v2f_t a = {1.0f, 1.0f};
  v2f_t b = {s, 0.0f};
  v8f_t c = {0.f, 0.f, 0.f, 0.f, 0.f, 0.f, 0.f, 0.f};
  c = __builtin_amdgcn_wmma_f32_16x16x4_f32(false, a, false, b, (short)0, c,
                                            false, false);
  float col = c[0];
  col += __shfl_xor(col, 1, 32);
  col += __shfl_xor(col, 2, 32);
  col += __shfl_xor(col, 4, 32);
  col += __shfl_xor(col, 8, 32);
  return col;
}

// Rotated box -> 4 corners (CCW order, matching the reference coefficient
// order [0.5,-0.5,-0.5,0.5] x [0.5,0.5,-0.5,-0.5]).
__device__ __forceinline__ void box_corners(float x, float y, float w, float h,
                                            float ang, float* cx, float* cy) {
  float sn, cs;
  sincosf(ang, &sn, &cs);
  float hw = 0.5f * w, hh = 0.5f * h;
  const float xs[4] = {hw, -hw, -hw, hw};
  const float ys[4] = {hh, hh, -hh, -hh};
#pragma unroll
  for (int k = 0; k < 4; ++k) {
    cx[k] = xs[k] * cs - ys[k] * sn + x;
    cy[k] = xs[k] * sn + ys[k] * cs + y;
  }
}

// Sutherland-Hodgman: clip CCW quad (c1) by the 4 half-planes of CCW quad
// (c2); shoelace the <=8-vertex result. Mathematically identical area to the
// reference's angular-sort construction for convex-convex intersection.
__device__ __forceinline__ float quad_intersection_area(const float* c1x,
                                                        const float* c1y,
                                                        const float* c2x,
                                                        const float* c2y) {
  float ax[8], ay[8], bx[8], by[8];
  int n = 4;
#pragma unroll
  for (int k = 0; k < 4; ++k) {
    ax[k] = c1x[k];
    ay[k] = c1y[k];
  }
  float* sx = ax;
  float* sy = ay;
  float* dx = bx;
  float* dy = by;
#pragma unroll
  for (int e = 0; e < 4; ++e) {
    float ex0 = c2x[e], ey0 = c2y[e];
    float ex1 = c2x[(e + 1) & 3], ey1 = c2y[(e + 1) & 3];
    float dex = ex1 - ex0, dey = ey1 - ey0;
    int m = 0;
    float lx = sx[n - 1], ly = sy[n - 1];
    float sideS = dex * (ly - ey0) - dey * (lx - ex0);  // >=0 == inside (CCW)
    for (int j = 0; j < n; ++j) {
      float cx = sx[j], cy = sy[j];
      float sideE = dex * (cy - ey0) - dey * (cx - ex0);
      if (sideE >= 0.0f) {
        if (sideS < 0.0f) {
          float tt = sideS / (sideS - sideE);
          dx[m] = lx + tt * (cx - lx);
          dy[m] = ly + tt * (cy - ly);
          ++m;
        }
        dx[m] = cx;
        dy[m] = cy;
        ++m;
      } else if (sideS >= 0.0f) {
        float tt = sideS / (sideS - sideE);
        dx[m] = lx + tt * (cx - lx);
        dy[m] = ly + tt * (cy - ly);
        ++m;
      }
      lx = cx;
      ly = cy;
      sideS = sideE;
    }
    n = m;
    if (n == 0) return 0.0f;
    float* t0 = sx; sx = dx; dx = t0;
    float* t1 = sy; sy = dy; dy = t1;
  }
  float area2 = 0.0f;
  float lx = sx[n - 1], ly = sy[n - 1];
  for (int j = 0; j < n; ++j) {
    area2 += lx * sy[j] - ly * sx[j];
    lx = sx[j];
    ly = sy[j];
  }
  return 0.5f * fabsf(area2);
}

// ---------------------------------------------------------------------------
// Pass 1: per-element CIoU loss * weight, deterministic block partial sums.
// ---------------------------------------------------------------------------
__global__ void __launch_bounds__(256)
ciou_partial_kernel(const float* __restrict__ pred,
                    const float* __restrict__ target,
                    const float* __restrict__ weight,
                    float* __restrict__ partials, int N) {
  const float EPS = 1e-6f;
  const float FOUR_OVER_PI2 = 0.4052847345693511f;  // 4 / pi^2
  float acc = 0.0f;
  int stride = gridDim.x * blockDim.x;
  for (int i = blockIdx.x * blockDim.x + threadIdx.x; i < N; i += stride) {
    const float* p = pred + 5 * i;
    const float* t = target + 5 * i;
    float px = p[0], py = p[1], pw = p[2], ph = p[3], pa = p[4];
    float tx = t[0], ty = t[1], tw = t[2], th = t[3], ta = t[4];

    float c1x[4], c1y[4], c2x[4], c2y[4];
    box_corners(px, py, pw, ph, pa, c1x, c1y);
    box_corners(tx, ty, tw, th, ta, c2x, c2y);

    float inter = quad_intersection_area(c1x, c1y, c2x, c2y);

    float a1 = pw * ph;
    float a2 = tw * th;
    float iou = fmaxf(inter / (a1 + a2 - inter), EPS);
    float ddx = px - tx, ddy = py - ty;
    float center_distance = ddx * ddx + ddy * ddy;
    float da = atanf(tw / (th + EPS)) - atanf(pw / (ph + EPS));
    float aspect = FOUR_OVER_PI2 * da * da;
    float v = aspect / (1.0f - iou + aspect);
    float alpha = v / (1.0f - iou + v);
    float loss = 1.0f - iou + center_distance / (pw * pw + ph * ph + EPS) +
                 alpha * v;
    acc += loss * weight[i];
  }

  // All control flow has reconverged: EXEC is all-ones for the WMMA.
  float wsum = wave_sum_wmma(acc);

  __shared__ float lred[8];
  int lane = threadIdx.x & 31;
  int wid = threadIdx.x >> 5;
  if (lane == 0) lred[wid] = wsum;
  __syncthreads();
  if (threadIdx.x == 0) {
    float s = 0.0f;
#pragma unroll
    for (int w = 0; w < 8; ++w) s += lred[w];
    partials[blockIdx.x] = s;
  }
}

// ---------------------------------------------------------------------------
// Pass 2: single block reduces the block partials; writes mean. Deterministic
// (fixed summation order per lane, WMMA+LDS tree), no float atomics.
// ---------------------------------------------------------------------------
__global__ void __launch_bounds__(256)
ciou_final_kernel(const float* __restrict__ partials, int nPartials, float invN,
                  float* __restrict__ out) {
  float acc = 0.0f;
  for (int i = threadIdx.x; i < nPartials; i += 256) acc += partials[i];

  float wsum = wave_sum_wmma(acc);

  __shared__ float lred[8];
  int lane = threadIdx.x & 31;
  int wid = threadIdx.x >> 5;
  if (lane == 0) lred[wid] = wsum;
  __syncthreads();
  if (threadIdx.x == 0) {
    float s = 0.0f;
#pragma unroll
    for (int w = 0; w < 8; ++w) s += lred[w];
    out[0] = s * invN;
  }
}

extern "C" void kernel_launch(void* const* d_in, const int* in_sizes, int n_in,
                              void* d_out, int out_size, void* d_ws,
                              size_t ws_size, hipStream_t stream) {
  const float* pred = (const float*)d_in[0];
  const float* target = (const float*)d_in[1];
  const float* weight = (const float*)d_in[2];
  float* out = (float*)d_out;
  float* partials = (float*)d_ws;
  int N = in_sizes[2];  // weight has one entry per box pair

  int blocks = 1024;
  size_t maxPartials = ws_size / sizeof(float);
  if ((size_t)blocks > maxPartials) blocks = (int)maxPartials;
  if (blocks < 1) blocks = 1;

  ciou_partial_kernel<<<blocks, 256, 0, stream>>>(pred, target, weight,
                                                  partials, N);
  ciou_final_kernel<<<1, 256, 0, stream>>>(partials, blocks,
                                           1.0f / (float)N, out);
}